// PointPillarScatter_44547400794349
// MI455X (gfx1250) — compile-verified
//
#include <hip/hip_runtime.h>
#include <hip/hip_bf16.h>
#include <stdint.h>

// Problem constants (match reference)
constexpr int NXc    = 512;
constexpr int NYc    = 512;
constexpr int Cc     = 64;
constexpr int Bc     = 4;
constexpr int CELLS  = NXc * NYc;      // nz == 1
constexpr int CELLS4 = CELLS / 4;      // 65536

// true HW vector type so __builtin_nontemporal_store/load accept it
typedef float v4f __attribute__((ext_vector_type(4)));
typedef int   v4i __attribute__((ext_vector_type(4)));

// ---------------------------------------------------------------------------
// Kernel 1: fill the pillar-id table with the sentinel 0 using the CDNA5
// async LDS->global store path (GLOBAL_STORE_ASYNC_FROM_LDS_B128, ASYNCcnt).
// Each lane stages 16B of zeros in LDS once, then fires fire-and-forget 16B
// async stores; the wave ends with s_wait_asynccnt 0.
// ---------------------------------------------------------------------------
__global__ void fill_table_async_kernel(int* __restrict__ table, unsigned n_chunks16)
{
    __shared__ int zbuf[1024];                       // 4 KB of zeros at LDS offset 0
    for (unsigned i = threadIdx.x; i < 1024u; i += blockDim.x) zbuf[i] = 0;
    __syncthreads();

    // per-lane LDS byte offset into the zero buffer (16B aligned, 4KB window)
    uint32_t lds_off = (threadIdx.x & 255u) * 16u;

    unsigned stride = gridDim.x * blockDim.x;
    for (unsigned i = blockIdx.x * blockDim.x + threadIdx.x; i < n_chunks16; i += stride) {
        uint64_t gaddr = (uint64_t)(uintptr_t)table + (uint64_t)i * 16u;
        asm volatile("global_store_async_from_lds_b128 %0, %1, off"
                     :: "v"(gaddr), "v"(lds_off)
                     : "memory");
    }
    asm volatile("s_wait_asynccnt 0x0" ::: "memory");
}

// ---------------------------------------------------------------------------
// Kernel 2: scatter pillar id (p+1) into the table at its (b, cell) slot.
// Slots are unique per the reference, so no atomics needed.
// ---------------------------------------------------------------------------
__global__ void build_table_kernel(const int* __restrict__ coords,
                                   int* __restrict__ table, int P)
{
    int p = blockIdx.x * blockDim.x + threadIdx.x;
    if (p < P) {
        int b   = coords[4 * p + 0];
        int idx = coords[4 * p + 1] + coords[4 * p + 2] * NXc + coords[4 * p + 3];
        table[b * CELLS + idx] = p + 1;
    }
}

// ---------------------------------------------------------------------------
// Kernel 3: gather. Thread owns 4 consecutive cells x 8 channels:
//   - one coalesced int4 table read serves 8 output channels (L2-resident)
//   - feature reads are two aligned float4 loads per live pillar (L2-resident)
//   - 8 fully coalesced NON-TEMPORAL float4 output stores: the 256MB output
//     is written once and bypasses L2 residency, preserving table/feat in L2.
// ---------------------------------------------------------------------------
__global__ void gather_kernel(const float* __restrict__ feat,
                              const int*  __restrict__ table,
                              float* __restrict__ out)
{
    unsigned tid   = blockIdx.x * blockDim.x + threadIdx.x;   // < B*8*CELLS4
    unsigned cell4 = tid & (CELLS4 - 1);                      // CELLS4 = 2^16
    unsigned g     = tid >> 16;                               // < B*8
    unsigned cg    = g & 7u;                                  // channel group (8 ch)
    unsigned b     = g >> 3;

    const v4i pid4 =
        reinterpret_cast<const v4i*>(table + (size_t)b * CELLS)[cell4];

    float f[4][8];
    int pids[4] = { pid4.x, pid4.y, pid4.z, pid4.w };
#pragma unroll
    for (int j = 0; j < 4; ++j) {
        int pid = pids[j];
        if (pid > 0) {
            const v4f* frow = reinterpret_cast<const v4f*>(
                feat + (size_t)(pid - 1) * Cc + cg * 8u);
            v4f a = frow[0];
            v4f c = frow[1];
            f[j][0] = a.x; f[j][1] = a.y; f[j][2] = a.z; f[j][3] = a.w;
            f[j][4] = c.x; f[j][5] = c.y; f[j][6] = c.z; f[j][7] = c.w;
        } else {
#pragma unroll
            for (int k = 0; k < 8; ++k) f[j][k] = 0.0f;
        }
    }

    v4f* out4 = reinterpret_cast<v4f*>(out);
#pragma unroll
    for (int c = 0; c < 8; ++c) {
        v4f v = { f[0][c], f[1][c], f[2][c], f[3][c] };
        __builtin_nontemporal_store(
            v, &out4[(size_t)(b * Cc + cg * 8u + c) * CELLS4 + cell4]);
    }
}

// ---------------------------------------------------------------------------
// Fallback path (only if ws_size < 4MB): plain zero-fill + direct scatter.
// ---------------------------------------------------------------------------
__global__ void zero_out_kernel(float* __restrict__ out, unsigned n4)
{
    unsigned i = blockIdx.x * blockDim.x + threadIdx.x;
    if (i < n4) {
        v4f z = { 0.f, 0.f, 0.f, 0.f };
        __builtin_nontemporal_store(z, &reinterpret_cast<v4f*>(out)[i]);
    }
}

__global__ void scatter_kernel(const float* __restrict__ feat,
                               const int*  __restrict__ coords,
                               float* __restrict__ out, int P)
{
    unsigned tid = blockIdx.x * blockDim.x + threadIdx.x;  // over P*C
    unsigned p = tid >> 6;                                 // C = 64
    unsigned c = tid & 63u;
    if ((int)p < P) {
        int b   = coords[4 * p + 0];
        int idx = coords[4 * p + 1] + coords[4 * p + 2] * NXc + coords[4 * p + 3];
        out[((size_t)(b * Cc + c)) * CELLS + idx] = feat[(size_t)p * Cc + c];
    }
}

// ---------------------------------------------------------------------------
extern "C" void kernel_launch(void* const* d_in, const int* in_sizes, int n_in,
                              void* d_out, int out_size, void* d_ws, size_t ws_size,
                              hipStream_t stream)
{
    const float* feat   = (const float*)d_in[0];
    const int*   coords = (const int*)d_in[1];
    float*       out    = (float*)d_out;
    const int    P      = in_sizes[0] / Cc;

    const size_t table_bytes = (size_t)Bc * CELLS * sizeof(int);   // 4 MB

    if (ws_size >= table_bytes) {
        int* table = (int*)d_ws;

        // 1) sentinel-fill the table via async LDS->global b128 stores
        unsigned n_chunks16 = (unsigned)(table_bytes / 16);        // 262144
        fill_table_async_kernel<<<1024, 256, 0, stream>>>(table, n_chunks16);

        // 2) scatter pillar ids
        build_table_kernel<<<(P + 255) / 256, 256, 0, stream>>>(coords, table, P);

        // 3) single coalesced gather pass over the whole output
        unsigned gather_threads = (unsigned)Bc * 8u * (unsigned)CELLS4; // 2,097,152
        gather_kernel<<<gather_threads / 256, 256, 0, stream>>>(feat, table, out);
    } else {
        // Fallback: zero + direct scatter
        unsigned n4 = (unsigned)(out_size / 4);                    // 16,777,216
        zero_out_kernel<<<(n4 + 255) / 256, 256, 0, stream>>>(out, n4);
        unsigned st = (unsigned)P * Cc;
        scatter_kernel<<<(st + 255) / 256, 256, 0, stream>>>(feat, coords, out, P);
    }
}